// Sinkhorn_16458314678942
// MI455X (gfx1250) — compile-verified
//
#include <hip/hip_runtime.h>
#include <math.h>

#define K_T   200
#define K_PAD 208
#define B_N   512
#define B_T   128
#define V_N   50000
#define V_C   16
#define N_CHUNK (V_N / V_C)   /* 3125 */
#define NVG   64
#define ALPHA_C 20.0f
#define THRESH  0.005f

#define U_STR  145   /* 128 + 17 : kills LDS bank conflicts for B-frag reads */
#define KM_STR 17
#define VL_STR 17

typedef float f32x2 __attribute__((ext_vector_type(2)));
typedef float f32x8 __attribute__((ext_vector_type(8)));

static __device__ __forceinline__ f32x8 wmma_f32(f32x2 a, f32x2 b, f32x8 c) {
  // V_WMMA_F32_16X16X4_F32 : D = A(16x4) * B(4x16) + C, fp32 throughout
  return __builtin_amdgcn_wmma_f32_16x16x4_f32(false, a, false, b, (short)0, c,
                                               false, false);
}

// ---------------------------------------------------------------- init
__global__ __launch_bounds__(256) void k_init(float* u, float* done) {
  int i = blockIdx.x * 256 + threadIdx.x;
  if (i < K_T * B_N) u[i] = 1.0f / (float)K_T;
  if (i == 0) *done = 0.0f;
}

// Kmat = exp(-ALPHA*(1-beta)) = exp(ALPHA*beta - ALPHA), layout (K,V) row-major
__global__ __launch_bounds__(256) void k_kmat(const float* __restrict__ beta,
                                              float* __restrict__ Km) {
  int i = blockIdx.x * 256 + threadIdx.x;
  if (i < K_T * V_N) Km[i] = expf(ALPHA_C * beta[i] - ALPHA_C);
}

// thetaT (K,B) from theta (B,K)
__global__ __launch_bounds__(256) void k_thetaT(const float* __restrict__ theta,
                                                float* __restrict__ thT) {
  int i = blockIdx.x * 256 + threadIdx.x;
  if (i < K_T * B_N) {
    int k = i >> 9, b = i & 511;
    thT[i] = theta[b * K_T + k];
  }
}

// per-document softmax stats over V
__global__ __launch_bounds__(256) void k_bowstats(const float* __restrict__ bow,
                                                  float* __restrict__ rmax,
                                                  float* __restrict__ rinv) {
  __shared__ float red[256];
  int b = blockIdx.x, t = threadIdx.x;
  const float* row = bow + (size_t)b * V_N;
  float m = -1e30f;
  for (int v = t; v < V_N; v += 256) m = fmaxf(m, row[v]);
  red[t] = m; __syncthreads();
  for (int s = 128; s > 0; s >>= 1) { if (t < s) red[t] = fmaxf(red[t], red[t + s]); __syncthreads(); }
  float M = red[0]; __syncthreads();
  float sum = 0.f;
  for (int v = t; v < V_N; v += 256) sum += expf(row[v] - M);
  red[t] = sum; __syncthreads();
  for (int s = 128; s > 0; s >>= 1) { if (t < s) red[t] += red[t + s]; __syncthreads(); }
  if (t == 0) { rmax[b] = M; rinv[b] = 1.0f / red[0]; }
}

// bowT (V,B) = softmax(bow,axis=1)^T via LDS-tiled transpose (coalesced both ways)
__global__ __launch_bounds__(256) void k_bowT(const float* __restrict__ bow,
                                              const float* __restrict__ rmax,
                                              const float* __restrict__ rinv,
                                              float* __restrict__ bowT) {
  __shared__ float tile[32][33];
  int v0 = blockIdx.x * 32, b0 = blockIdx.y * 32;
  int t = threadIdx.x;
  for (int idx = t; idx < 1024; idx += 256) {
    int r = idx >> 5, c = idx & 31;              // r: b-local, c: v-local
    int v = v0 + c, b = b0 + r;
    float x = 0.0f;
    if (v < V_N) x = expf(bow[(size_t)b * V_N + v] - rmax[b]) * rinv[b];
    tile[r][c] = x;
  }
  __syncthreads();
  for (int idx = t; idx < 1024; idx += 256) {
    int r = idx >> 5, c = idx & 31;              // r: v-local, c: b-local
    int v = v0 + r;
    if (v < V_N) bowT[(size_t)v * B_N + b0 + c] = tile[c][r];
  }
}

// ---------------------------------------------------------------- fused iteration
// S = Kmat^T@u ; _v = bowT/S ; Tpart[gx] += Kmat@_v
// Double-buffered Kmat tiles: next chunk prefetched into registers while the
// current chunk's ~102 WMMAs execute; one barrier per chunk.
__global__ __launch_bounds__(256) void k_iter(const float* __restrict__ Km,
                                              const float* __restrict__ bowT,
                                              const float* __restrict__ u,
                                              float* __restrict__ Tpart,
                                              float* __restrict__ vout,
                                              const float* __restrict__ done,
                                              int write_v) {
  if (*done >= 0.5f) return;                     // freeze semantics (uniform)
  __shared__ float u_lds[K_PAD * U_STR];         // 120,640 B
  __shared__ float km_lds[2][K_PAD * KM_STR];    //  28,288 B (double buffered)
  __shared__ float v_lds[8 * 16 * VL_STR];       //   8,704 B
  int t = threadIdx.x, lane = t & 31, wave = t >> 5;
  int half = lane >> 4, l16 = lane & 15;
  int cb0 = blockIdx.y * B_T;
  int colw = wave * 16;
  int col = cb0 + colw + l16;

  for (int idx = t; idx < K_PAD * B_T; idx += 256) {
    int k = idx >> 7, c = idx & 127;
    u_lds[k * U_STR + c] = (k < K_T) ? u[k * B_N + cb0 + c] : 0.0f;
  }
  for (int idx = t; idx < (K_PAD - K_T) * V_C; idx += 256) {
    int o = (K_T + (idx >> 4)) * KM_STR + (idx & 15);
    km_lds[0][o] = 0.0f; km_lds[1][o] = 0.0f;
  }
  { // stage first chunk into buffer 0
    int v0 = blockIdx.x * V_C;
    for (int idx = t; idx < K_T * V_C; idx += 256)
      km_lds[0][(idx >> 4) * KM_STR + (idx & 15)] = Km[(idx >> 4) * V_N + v0 + (idx & 15)];
  }
  __syncthreads();

  f32x8 zero = {0.f, 0.f, 0.f, 0.f, 0.f, 0.f, 0.f, 0.f};
  f32x8 acc[13];
#pragma unroll
  for (int m = 0; m < 13; ++m) acc[m] = zero;
  float* vw = &v_lds[wave * 16 * VL_STR];

  int p = 0;
  for (int ch = blockIdx.x; ch < N_CHUNK; ch += NVG, p ^= 1) {
    int v0 = ch * V_C;
    const float* km = km_lds[p];

    // issue current-chunk bowT loads early (covered by the 50-WMMA S loop)
    float bw[8];
#pragma unroll
    for (int r = 0; r < 8; ++r)
      bw[r] = bowT[(size_t)(v0 + r + half * 8) * B_N + col];

    // issue next-chunk Kmat loads into registers (stored to other buffer later)
    int nch = ch + NVG;
    float pre[13];
    if (nch < N_CHUNK) {
      int nv0 = nch * V_C;
#pragma unroll
      for (int j = 0; j < 13; ++j) {
        int idx = t + j * 256;
        if (idx < K_T * V_C)
          pre[j] = Km[(idx >> 4) * V_N + nv0 + (idx & 15)];
      }
    }

    // S frag (16 v-rows x 16 cols), inner dim K=200 (50 steps of 4)
    f32x8 s = zero;
#pragma unroll 5
    for (int kk = 0; kk < 50; ++kk) {
      int kb = kk * 4 + half * 2;
      f32x2 a, b;
      a[0] = km[kb * KM_STR + l16];
      a[1] = km[(kb + 1) * KM_STR + l16];
      b[0] = u_lds[kb * U_STR + colw + l16];
      b[1] = u_lds[(kb + 1) * U_STR + colw + l16];
      s = wmma_f32(a, b, s);
    }
    // _v = bowT * rcp(S) : v_rcp_f32 instead of the ~12-op IEEE divide sequence
    float vv[8];
#pragma unroll
    for (int r = 0; r < 8; ++r) {
      vv[r] = bw[r] * __builtin_amdgcn_rcpf(s[r]);
      vw[(r + half * 8) * VL_STR + l16] = vv[r];   // LDS in-order per wave
    }
    if (write_v) {
#pragma unroll
      for (int r = 0; r < 8; ++r)
        vout[(size_t)(v0 + r + half * 8) * B_N + col] = vv[r];
    }
    // T += Km_chunk(200x16) @ _v(16x16cols) : 13 M-tiles x 4 k-steps
#pragma unroll
    for (int st = 0; st < 4; ++st) {
      int vr = st * 4 + half * 2;
      f32x2 b2;
      b2[0] = vw[vr * VL_STR + l16];
      b2[1] = vw[(vr + 1) * VL_STR + l16];
#pragma unroll
      for (int m = 0; m < 13; ++m) {
        f32x2 a2;
        a2[0] = km[(m * 16 + l16) * KM_STR + vr];
        a2[1] = km[(m * 16 + l16) * KM_STR + vr + 1];
        acc[m] = wmma_f32(a2, b2, acc[m]);
      }
    }
    // commit prefetched next chunk into the other buffer
    if (nch < N_CHUNK) {
      float* kn = km_lds[p ^ 1];
#pragma unroll
      for (int j = 0; j < 13; ++j) {
        int idx = t + j * 256;
        if (idx < K_T * V_C)
          kn[(idx >> 4) * KM_STR + (idx & 15)] = pre[j];
      }
    }
    __syncthreads();   // single barrier per chunk
  }
  float* Tp = Tpart + (size_t)blockIdx.x * (K_T * B_N);   // deterministic partials
#pragma unroll
  for (int m = 0; m < 13; ++m)
#pragma unroll
    for (int r = 0; r < 8; ++r) {
      int krow = m * 16 + r + half * 8;
      if (krow < K_T) Tp[krow * B_N + col] = acc[m][r];
    }
}

// u = thetaT / sum_g Tpart[g]
__global__ __launch_bounds__(256) void k_finish(const float* __restrict__ thT,
                                                const float* __restrict__ Tpart,
                                                float* __restrict__ u,
                                                const float* __restrict__ done) {
  if (*done >= 0.5f) return;
  int i = blockIdx.x * 256 + threadIdx.x;
  if (i < K_T * B_N) {
    float sum = 0.f;
#pragma unroll 4
    for (int g = 0; g < NVG; ++g) sum += Tpart[(size_t)g * (K_T * B_N) + i];
    u[i] = thT[i] / sum;
  }
}

// errpart[gx][b] = sum_v |vn * (K^T@un2) - bowT| over this group's chunks
__global__ __launch_bounds__(256) void k_err(const float* __restrict__ Km,
                                             const float* __restrict__ bowT,
                                             const float* __restrict__ u,
                                             const float* __restrict__ vbuf,
                                             float* __restrict__ errpart,
                                             const float* __restrict__ done) {
  if (*done >= 0.5f) return;
  __shared__ float u_lds[K_PAD * U_STR];
  __shared__ float km_lds[2][K_PAD * KM_STR];
  int t = threadIdx.x, lane = t & 31, wave = t >> 5;
  int half = lane >> 4, l16 = lane & 15;
  int cb0 = blockIdx.y * B_T;
  int colw = wave * 16;
  int col = cb0 + colw + l16;

  for (int idx = t; idx < K_PAD * B_T; idx += 256) {
    int k = idx >> 7, c = idx & 127;
    u_lds[k * U_STR + c] = (k < K_T) ? u[k * B_N + cb0 + c] : 0.0f;
  }
  { // stage first chunk
    int v0 = blockIdx.x * V_C;
    for (int idx = t; idx < K_T * V_C; idx += 256)
      km_lds[0][(idx >> 4) * KM_STR + (idx & 15)] = Km[(idx >> 4) * V_N + v0 + (idx & 15)];
  }
  __syncthreads();

  f32x8 zero = {0.f, 0.f, 0.f, 0.f, 0.f, 0.f, 0.f, 0.f};
  float eacc = 0.f;
  int p = 0;
  for (int ch = blockIdx.x; ch < N_CHUNK; ch += NVG, p ^= 1) {
    int v0 = ch * V_C;
    const float* km = km_lds[p];
    float bw[8], vb[8];
#pragma unroll
    for (int r = 0; r < 8; ++r) {
      size_t g = (size_t)(v0 + r + half * 8) * B_N + col;
      bw[r] = bowT[g];
      vb[r] = vbuf[g];
    }
    int nch = ch + NVG;
    float pre[13];
    if (nch < N_CHUNK) {
      int nv0 = nch * V_C;
#pragma unroll
      for (int j = 0; j < 13; ++j) {
        int idx = t + j * 256;
        if (idx < K_T * V_C)
          pre[j] = Km[(idx >> 4) * V_N + nv0 + (idx & 15)];
      }
    }
    f32x8 s = zero;
#pragma unroll 5
    for (int kk = 0; kk < 50; ++kk) {
      int kb = kk * 4 + half * 2;
      f32x2 a, b;
      a[0] = km[kb * KM_STR + l16];
      a[1] = km[(kb + 1) * KM_STR + l16];
      b[0] = u_lds[kb * U_STR + colw + l16];
      b[1] = u_lds[(kb + 1) * U_STR + colw + l16];
      s = wmma_f32(a, b, s);
    }
#pragma unroll
    for (int r = 0; r < 8; ++r) eacc += fabsf(vb[r] * s[r] - bw[r]);
    if (nch < N_CHUNK) {
      float* kn = km_lds[p ^ 1];
#pragma unroll
      for (int j = 0; j < 13; ++j) {
        int idx = t + j * 256;
        if (idx < K_T * V_C)
          kn[(idx >> 4) * KM_STR + (idx & 15)] = pre[j];
      }
    }
    __syncthreads();
  }
  float o = __shfl_down(eacc, 16, 32);           // combine the two half-rows
  if (half == 0) errpart[blockIdx.x * B_N + col] = eacc + o;
}

__global__ __launch_bounds__(512) void k_done(const float* __restrict__ errpart,
                                              float* __restrict__ done) {
  __shared__ float red[512];
  int t = threadIdx.x;
  float e = 0.f;
  for (int g = 0; g < NVG; ++g) e += errpart[g * B_N + t];
  red[t] = e; __syncthreads();
  for (int s = 256; s > 0; s >>= 1) { if (t < s) red[t] = fmaxf(red[t], red[t + s]); __syncthreads(); }
  if (t == 0 && red[0] <= THRESH) *done = 1.0f;
}

// Tpart[gx] = (Kmat*M) @ v  with KM = -K*ln(K)/ALPHA computed at LDS-store time
__global__ __launch_bounds__(256) void k_div(const float* __restrict__ Km,
                                             const float* __restrict__ vbuf,
                                             float* __restrict__ Tpart) {
  __shared__ float km_lds[2][K_PAD * KM_STR];
  __shared__ float v_lds[8 * 16 * VL_STR];
  int t = threadIdx.x, lane = t & 31, wave = t >> 5;
  int half = lane >> 4, l16 = lane & 15;
  int cb0 = blockIdx.y * B_T;
  int colw = wave * 16;
  int col = cb0 + colw + l16;

  for (int idx = t; idx < (K_PAD - K_T) * V_C; idx += 256) {
    int o = (K_T + (idx >> 4)) * KM_STR + (idx & 15);
    km_lds[0][o] = 0.0f; km_lds[1][o] = 0.0f;
  }
  { // stage first chunk (transformed)
    int v0 = blockIdx.x * V_C;
    for (int idx = t; idx < K_T * V_C; idx += 256) {
      float kv = Km[(idx >> 4) * V_N + v0 + (idx & 15)];
      km_lds[0][(idx >> 4) * KM_STR + (idx & 15)] = -kv * logf(kv) * (1.0f / ALPHA_C);
    }
  }
  // preload v columns for first chunk
  float vb[8];
  {
    int v0 = blockIdx.x * V_C;
#pragma unroll
    for (int r = 0; r < 8; ++r)
      vb[r] = vbuf[(size_t)(v0 + r + half * 8) * B_N + col];
  }
  __syncthreads();

  f32x8 zero = {0.f, 0.f, 0.f, 0.f, 0.f, 0.f, 0.f, 0.f};
  f32x8 acc[13];
#pragma unroll
  for (int m = 0; m < 13; ++m) acc[m] = zero;
  float* vw = &v_lds[wave * 16 * VL_STR];

  int p = 0;
  for (int ch = blockIdx.x; ch < N_CHUNK; ch += NVG, p ^= 1) {
    const float* km = km_lds[p];
    // commit this chunk's v (in-order LDS per wave: T-phase reads are safe)
#pragma unroll
    for (int r = 0; r < 8; ++r) vw[(r + half * 8) * VL_STR + l16] = vb[r];

    int nch = ch + NVG;
    float pre[13], vbn[8];
    if (nch < N_CHUNK) {
      int nv0 = nch * V_C;
#pragma unroll
      for (int j = 0; j < 13; ++j) {
        int idx = t + j * 256;
        if (idx < K_T * V_C)
          pre[j] = Km[(idx >> 4) * V_N + nv0 + (idx & 15)];
      }
#pragma unroll
      for (int r = 0; r < 8; ++r)
        vbn[r] = vbuf[(size_t)(nv0 + r + half * 8) * B_N + col];
    }
#pragma unroll
    for (int st = 0; st < 4; ++st) {
      int vr = st * 4 + half * 2;
      f32x2 b2;
      b2[0] = vw[vr * VL_STR + l16];
      b2[1] = vw[(vr + 1) * VL_STR + l16];
#pragma unroll
      for (int m = 0; m < 13; ++m) {
        f32x2 a2;
        a2[0] = km[(m * 16 + l16) * KM_STR + vr];
        a2[1] = km[(m * 16 + l16) * KM_STR + vr + 1];
        acc[m] = wmma_f32(a2, b2, acc[m]);
      }
    }
    if (nch < N_CHUNK) {
      float* kn = km_lds[p ^ 1];
#pragma unroll
      for (int j = 0; j < 13; ++j) {
        int idx = t + j * 256;
        if (idx < K_T * V_C) {
          float kv = pre[j];
          kn[(idx >> 4) * KM_STR + (idx & 15)] = -kv * logf(kv) * (1.0f / ALPHA_C);
        }
      }
#pragma unroll
      for (int r = 0; r < 8; ++r) vb[r] = vbn[r];
    }
    __syncthreads();
  }
  float* Tp = Tpart + (size_t)blockIdx.x * (K_T * B_N);
#pragma unroll
  for (int m = 0; m < 13; ++m)
#pragma unroll
    for (int r = 0; r < 8; ++r) {
      int krow = m * 16 + r + half * 8;
      if (krow < K_T) Tp[krow * B_N + col] = acc[m][r];
    }
}

__global__ __launch_bounds__(256) void k_divfinal(const float* __restrict__ u,
                                                  const float* __restrict__ Tpart,
                                                  float* __restrict__ out) {
  __shared__ float red[256];
  int t = threadIdx.x;
  float s = 0.f;
  for (int i = t; i < K_T * B_N; i += 256) {
    float g = 0.f;
    for (int gg = 0; gg < NVG; ++gg) g += Tpart[(size_t)gg * (K_T * B_N) + i];
    s += u[i] * g;
  }
  red[t] = s; __syncthreads();
  for (int q = 128; q > 0; q >>= 1) { if (t < q) red[t] += red[t + q]; __syncthreads(); }
  if (t == 0) out[0] = red[0] / (float)B_N;
}

// ----------------------------------------------------------------
extern "C" void kernel_launch(void* const* d_in, const int* in_sizes, int n_in,
                              void* d_out, int out_size, void* d_ws, size_t ws_size,
                              hipStream_t stream) {
  (void)in_sizes; (void)n_in; (void)out_size; (void)ws_size;
  const float* beta  = (const float*)d_in[0];
  const float* theta = (const float*)d_in[1];
  const float* bow   = (const float*)d_in[2];
  float* out = (float*)d_out;

  float* ws = (float*)d_ws;
  float* Km    = ws; ws += (size_t)K_T * V_N;        // 40.0 MB
  float* bowT  = ws; ws += (size_t)V_N * B_N;        // 102.4 MB
  float* vbuf  = ws; ws += (size_t)V_N * B_N;        // 102.4 MB
  float* Tpart = ws; ws += (size_t)NVG * K_T * B_N;  // 26.2 MB
  float* u     = ws; ws += K_T * B_N;
  float* thT   = ws; ws += K_T * B_N;
  float* errp  = ws; ws += NVG * B_N;
  float* rmax  = ws; ws += B_N;
  float* rinv  = ws; ws += B_N;
  float* done  = ws; ws += 1;

  k_init  <<<(K_T * B_N + 255) / 256, 256, 0, stream>>>(u, done);
  k_kmat  <<<(K_T * V_N + 255) / 256, 256, 0, stream>>>(beta, Km);
  k_thetaT<<<(K_T * B_N + 255) / 256, 256, 0, stream>>>(theta, thT);
  k_bowstats<<<B_N, 256, 0, stream>>>(bow, rmax, rinv);
  k_bowT  <<<dim3((V_N + 31) / 32, B_N / 32), 256, 0, stream>>>(bow, rmax, rinv, bowT);

  for (int blk = 0; blk < 5; ++blk) {
    for (int it = 0; it <= 20; ++it) {               // 20 plain + 1 check update
      k_iter  <<<dim3(NVG, B_N / B_T), 256, 0, stream>>>(Km, bowT, u, Tpart, vbuf,
                                                         done, it == 20 ? 1 : 0);
      k_finish<<<(K_T * B_N + 255) / 256, 256, 0, stream>>>(thT, Tpart, u, done);
    }
    k_err <<<dim3(NVG, B_N / B_T), 256, 0, stream>>>(Km, bowT, u, vbuf, errp, done);
    k_done<<<1, 512, 0, stream>>>(errp, done);
  }
  k_div     <<<dim3(NVG, B_N / B_T), 256, 0, stream>>>(Km, vbuf, Tpart);
  k_divfinal<<<1, 256, 0, stream>>>(u, Tpart, out);
}